// KabschRMSD_89988154786048
// MI455X (gfx1250) — compile-verified
//
#include <hip/hip_runtime.h>
#include <hip/hip_bf16.h>

typedef float v2f __attribute__((ext_vector_type(2)));
typedef float v8f __attribute__((ext_vector_type(8)));

// ---------------------------------------------------------------------------
// Kernel 1: per-ligand raw moments via WMMA (4 ligands per wave, block-diag
// 4x4 tiles inside one 16x16 f32 WMMA accumulator), then closed-form Kabsch
// RMSD per ligand written to workspace.
//
// Tile layout (per wave, batches b0..b0+3):
//   A (16x4)  row M = 4*m + i : i<3 -> P[b0+m][n][i], i==3 -> 1.0
//   B (4x16)  col N = 4*m + j : j<3 -> Q[b0+m][n][j], j==3 -> 1.0
//   D[4m+i][4m+j] accumulates over n:
//     i<3,j<3 : sum P_i Q_j      (raw cross-covariance)
//     i<3,j=3 : sum P_i          (for Pm)
//     i=3,j<3 : sum Q_j          (for Qm)
//     i=3,j=3 : N
// ---------------------------------------------------------------------------
__global__ void __launch_bounds__(256)
kabsch_rmsd_wmma_kernel(const float* __restrict__ P,
                        const float* __restrict__ Q,
                        float* __restrict__ rmsd_out,
                        int B)
{
    __shared__ float lds[8][32][10]; // [wave][lane][{c0..c7, sp, sq}]

    const int wave = threadIdx.x >> 5;
    const int lane = threadIdx.x & 31;
    const int gwave = blockIdx.x * 8 + wave;
    const int b0 = gwave * 4;

    const int half = lane >> 4;      // 0: K = n,n+1 ; 1: K = n+2,n+3
    const int l    = lane & 15;
    const int m    = l >> 2;         // ligand-in-tile 0..3
    const int sub  = l & 3;          // coord index (3 => constant-ones row/col)

    int batch = b0 + m; if (batch >= B) batch = B - 1;  // safety clamp
    const int  subc = (sub < 3) ? sub : 0;
    const bool real = (sub < 3);
    const float msk = real ? 1.0f : 0.0f;

    const float* pbase = P + (size_t)batch * 384 + subc;  // 128*3 floats/ligand
    const float* qbase = Q + (size_t)batch * 384 + subc;

    v8f c = {0.f,0.f,0.f,0.f,0.f,0.f,0.f,0.f};
    float sp = 0.0f, sq = 0.0f;
    const int k0 = 2 * half;

    #pragma unroll 4
    for (int n = 0; n < 128; n += 4) {
        float a0 = pbase[(n + k0)     * 3];
        float a1 = pbase[(n + k0 + 1) * 3];
        float b0v = qbase[(n + k0)     * 3];
        float b1v = qbase[(n + k0 + 1) * 3];
        a0  = real ? a0  : 1.0f;
        a1  = real ? a1  : 1.0f;
        b0v = real ? b0v : 1.0f;
        b1v = real ? b1v : 1.0f;
        sp = fmaf(msk, a0 * a0 + a1 * a1, sp);
        sq = fmaf(msk, b0v * b0v + b1v * b1v, sq);
        v2f av = {a0, a1};
        v2f bv = {b0v, b1v};
        // D(16x16,f32) += A(16x4,f32) * B(4x16,f32)
        c = __builtin_amdgcn_wmma_f32_16x16x4_f32(
                false, av, false, bv, (short)0, c, false, false);
    }

    // spill accumulators + partial norms to LDS for the solver phase
    #pragma unroll
    for (int r = 0; r < 8; ++r) lds[wave][lane][r] = c[r];
    lds[wave][lane][8] = sp;
    lds[wave][lane][9] = sq;
    __syncthreads();

    // -------- per-ligand finish: lanes 0..3 of each wave --------
    if (lane < 4) {
        const int mm = lane;
        const int bb = b0 + mm;
        if (bb < B) {
            // gather this ligand's 4x4 diagonal block of D
            // D[M][Ncol] lives at lane = Ncol + 16*(M>=8), reg = M & 7
            float Dm[4][4];
            #pragma unroll
            for (int i = 0; i < 4; ++i) {
                const int M  = 4 * mm + i;
                const int hb = (M >= 8) ? 16 : 0;
                const int r  = M & 7;
                #pragma unroll
                for (int j = 0; j < 4; ++j)
                    Dm[i][j] = lds[wave][4 * mm + j + hb][r];
            }
            // raw squared norms: P row (mm,i) lives in lanes 4mm+i and +16;
            // Q col (mm,j) likewise.
            float GAr = 0.0f, GBr = 0.0f;
            #pragma unroll
            for (int i = 0; i < 3; ++i) {
                GAr += lds[wave][4 * mm + i][8] + lds[wave][4 * mm + i + 16][8];
                GBr += lds[wave][4 * mm + i][9] + lds[wave][4 * mm + i + 16][9];
            }

            const float Nf = 128.0f, invN = 1.0f / 128.0f;
            float Pm[3], Qm[3];
            #pragma unroll
            for (int i = 0; i < 3; ++i) { Pm[i] = Dm[i][3] * invN; Qm[i] = Dm[3][i] * invN; }

            float A[3][3];
            #pragma unroll
            for (int i = 0; i < 3; ++i)
                #pragma unroll
                for (int j = 0; j < 3; ++j)
                    A[i][j] = Dm[i][j] - Nf * Pm[i] * Qm[j];

            const float GA = GAr - Nf * (Pm[0]*Pm[0] + Pm[1]*Pm[1] + Pm[2]*Pm[2]);
            const float GB = GBr - Nf * (Qm[0]*Qm[0] + Qm[1]*Qm[1] + Qm[2]*Qm[2]);

            const float det =
                A[0][0]*(A[1][1]*A[2][2] - A[1][2]*A[2][1]) -
                A[0][1]*(A[1][0]*A[2][2] - A[1][2]*A[2][0]) +
                A[0][2]*(A[1][0]*A[2][1] - A[1][1]*A[2][0]);
            const float dsgn = (det > 0.0f) ? 1.0f : ((det < 0.0f) ? -1.0f : 0.0f);

            // M = A^T A (symmetric PSD); closed-form eigenvalues
            float M00=0,M01=0,M02=0,M11=0,M12=0,M22=0;
            #pragma unroll
            for (int k = 0; k < 3; ++k) {
                M00 += A[k][0]*A[k][0]; M01 += A[k][0]*A[k][1];
                M02 += A[k][0]*A[k][2]; M11 += A[k][1]*A[k][1];
                M12 += A[k][1]*A[k][2]; M22 += A[k][2]*A[k][2];
            }
            const float q  = (M00 + M11 + M22) * (1.0f/3.0f);
            const float p1 = M01*M01 + M02*M02 + M12*M12;
            const float p2 = (M00-q)*(M00-q) + (M11-q)*(M11-q) + (M22-q)*(M22-q) + 2.0f*p1;
            float e1, e2, e3;
            if (p2 < 1e-30f) {
                e1 = e2 = e3 = q;
            } else {
                const float p  = sqrtf(p2 * (1.0f/6.0f));
                const float ip = 1.0f / p;
                const float b00=(M00-q)*ip, b11=(M11-q)*ip, b22=(M22-q)*ip;
                const float b01=M01*ip, b02=M02*ip, b12=M12*ip;
                float r = 0.5f * (b00*(b11*b22 - b12*b12)
                                - b01*(b01*b22 - b12*b02)
                                + b02*(b01*b12 - b11*b02));
                r = fminf(1.0f, fmaxf(-1.0f, r));
                const float phi = acosf(r) * (1.0f/3.0f);
                e1 = q + 2.0f*p*cosf(phi);
                e3 = q + 2.0f*p*cosf(phi + 2.0943951023931953f); // +2pi/3 (smallest)
                e2 = 3.0f*q - e1 - e3;
            }
            const float s1 = sqrtf(fmaxf(e1, 0.0f));
            const float s2 = sqrtf(fmaxf(e2, 0.0f));
            const float s3 = sqrtf(fmaxf(e3, 0.0f));

            const float msd = fmaxf(0.0f, (GA + GB - 2.0f*(s1 + s2 + dsgn*s3)) * invN);
            rmsd_out[bb] = sqrtf(msd);
        }
    }
}

// ---------------------------------------------------------------------------
// Kernel 2: deterministic mean over B values (fixed-order strided + LDS tree)
// ---------------------------------------------------------------------------
__global__ void __launch_bounds__(256)
mean_reduce_kernel(const float* __restrict__ v, float* __restrict__ out, int n)
{
    __shared__ float s[256];
    float acc = 0.0f;
    for (int i = threadIdx.x; i < n; i += 256) acc += v[i];
    s[threadIdx.x] = acc;
    __syncthreads();
    #pragma unroll
    for (int off = 128; off > 0; off >>= 1) {
        if (threadIdx.x < off) s[threadIdx.x] += s[threadIdx.x + off];
        __syncthreads();
    }
    if (threadIdx.x == 0) out[0] = s[0] / (float)n;
}

extern "C" void kernel_launch(void* const* d_in, const int* in_sizes, int n_in,
                              void* d_out, int out_size, void* d_ws, size_t ws_size,
                              hipStream_t stream)
{
    const float* P = (const float*)d_in[0];   // ligs_coords_pred [B,128,3] f32
    const float* Q = (const float*)d_in[1];   // ligs_coords      [B,128,3] f32
    float* out = (float*)d_out;               // scalar mean rmsd
    float* ws  = (float*)d_ws;                // per-ligand rmsd, B floats

    const int B = in_sizes[0] / 384;          // 128*3 floats per ligand
    const int waves  = (B + 3) / 4;           // 4 ligands per wave
    const int blocks = (waves + 7) / 8;       // 8 waves (256 thr) per block

    kabsch_rmsd_wmma_kernel<<<blocks, 256, 0, stream>>>(P, Q, ws, B);
    mean_reduce_kernel<<<1, 256, 0, stream>>>(ws, out, B);
}